// GridSamplePScan_12996571037932
// MI455X (gfx1250) — compile-verified
//
#include <hip/hip_runtime.h>
#include <hip/hip_bf16.h>

typedef __attribute__((ext_vector_type(2))) float v2f;
typedef __attribute__((ext_vector_type(8))) float v8f;

#define Lt 16
#define Cc 32
#define HID 16
#define HW  4096   // 64*64

// ---------------------------------------------------------------------------
// Kernel 1: cum[t] = cumsum over t of flows[t]  (L=16, 2 comps, 4096 pixels)
// ---------------------------------------------------------------------------
__global__ void cumsum_kernel(const float* __restrict__ flows,
                              float* __restrict__ cum) {
    int pix = blockIdx.x * blockDim.x + threadIdx.x;  // 0..4095
    float ax = 0.0f, ay = 0.0f;
    #pragma unroll
    for (int t = 0; t < Lt; ++t) {
        ax += flows[(t * 2 + 0) * HW + pix];
        ay += flows[(t * 2 + 1) * HW + pix];
        cum[(t * 2 + 0) * HW + pix] = ax;
        cum[(t * 2 + 1) * HW + pix] = ay;
    }
}

// ---------------------------------------------------------------------------
// Kernel 2: factored first-layer projections via V_WMMA_F32_16X16X4_F32.
//   A[t] = w1[:, 0:32]  @ images[t] + b1      (16 x 4096 per t)
//   B[k] = w1[:,32:64]  @ images[k]           (16 x 4096 per k)
// One wave computes one 16x16 output tile; 8 WMMA steps of K=4 cover C=32.
// ISA 16x16x4 f32 layout: lanes 0-15 hold K={0,1}/M rows, lanes 16-31 K={2,3};
// C/D: lane%16 = N, vgpr i = M=i (+8 for lanes 16-31).
// ---------------------------------------------------------------------------
__global__ void proj_kernel(const float* __restrict__ images,
                            const float* __restrict__ w1,
                            const float* __restrict__ b1,
                            float* __restrict__ Aout,
                            float* __restrict__ Bout) {
    int wave = (blockIdx.x * blockDim.x + threadIdx.x) >> 5;  // 0..8191
    int lane = threadIdx.x & 31;
    int j       = wave >> 12;        // 0: A-part (w1[:,0:32]),  1: B-part
    int rem     = wave & 4095;
    int t       = rem >> 8;          // frame 0..15
    int colBase = (rem & 255) << 4;  // pixel tile base, 256 tiles of 16
    int hi      = lane >> 4;         // 0 or 1 (lane half)
    int row     = lane & 15;

    const float* img = images + t * Cc * HW;
    const float* W   = w1 + j * Cc;           // row stride 64 in w1
    v8f acc = {};
    #pragma unroll
    for (int kk = 0; kk < 8; ++kk) {
        int c0 = kk * 4 + hi * 2;             // channel of K-slot 0 for this half
        v2f a, b;
        a.x = W[row * 64 + c0];               // A[M=row][K=c0]
        a.y = W[row * 64 + c0 + 1];           // A[M=row][K=c0+1]
        b.x = img[(c0    ) * HW + colBase + row];  // B[K=c0][N=row]
        b.y = img[(c0 + 1) * HW + colBase + row];  // B[K=c0+1][N=row]
        acc = __builtin_amdgcn_wmma_f32_16x16x4_f32(
                  /*neg_a=*/false, a, /*neg_b=*/false, b,
                  /*c_mod=*/(short)0, acc, /*reuse_a=*/false, /*reuse_b=*/false);
    }
    // Branch-free epilogue: unconditional b1 load, scaled by uniform jscale
    // (avoids the 8 divergent exec-save blocks seen in the previous build).
    float jscale = (j == 0) ? 1.0f : 0.0f;
    float* out = (j == 0) ? Aout : Bout;
    #pragma unroll
    for (int i = 0; i < 8; ++i) {
        int m = i + 8 * hi;                   // output row (hidden unit)
        out[(t * HID + m) * HW + colBase + row] = acc[i] + jscale * b1[m];
    }
}

// ---------------------------------------------------------------------------
// Kernel 3: fused residual-MLP + grid wrap + bilinear gather + decay pscan.
// One thread per (t, pixel). P_A[t] is k-invariant -> cached in registers.
// ---------------------------------------------------------------------------
__global__ void fuse_kernel(const float* __restrict__ images,
                            const float* __restrict__ cum,
                            const float* __restrict__ Ain,
                            const float* __restrict__ Bin,
                            const float* __restrict__ decay_log,
                            const float* __restrict__ w2,
                            const float* __restrict__ b2,
                            float* __restrict__ out) {
    int t   = blockIdx.y;
    int pix = blockIdx.x * blockDim.x + threadIdx.x;
    int x = pix & 63, y = pix >> 6;
    float gxb = (2.0f * (float)x + 1.0f) * (1.0f / 64.0f) - 1.0f;
    float gyb = (2.0f * (float)y + 1.0f) * (1.0f / 64.0f) - 1.0f;
    float decay = expf(decay_log[0]);

    // w2/b2 are uniform -> compiler keeps these in SGPRs (SMEM loads)
    float w2r0[HID], w2r1[HID];
    #pragma unroll
    for (int m = 0; m < HID; ++m) { w2r0[m] = w2[m]; w2r1[m] = w2[HID + m]; }
    float b20 = b2[0], b21 = b2[1];

    float Areg[HID];
    #pragma unroll
    for (int m = 0; m < HID; ++m) Areg[m] = Ain[(t * HID + m) * HW + pix];
    float ctx = cum[(t * 2 + 0) * HW + pix];
    float cty = cum[(t * 2 + 1) * HW + pix];

    float acc[Cc];
    #pragma unroll
    for (int c = 0; c < Cc; ++c) acc[c] = 0.0f;

    for (int k = 0; k <= t; ++k) {
        float wt = expf(-decay * (float)(t - k));
        // residual MLP: h = relu(A_t + B_k), res = w2 @ h + b2
        float r0 = b20, r1 = b21;
        #pragma unroll
        for (int m = 0; m < HID; ++m) {
            float h = Areg[m] + Bin[(k * HID + m) * HW + pix];
            h = fmaxf(h, 0.0f);
            r0 += w2r0[m] * h;
            r1 += w2r1[m] * h;
        }
        float gx = gxb + (ctx - cum[(k * 2 + 0) * HW + pix]) + r0;
        float gy = gyb + (cty - cum[(k * 2 + 1) * HW + pix]) + r1;
        // wrap x: mod(gx+1, 2) - 1  (non-negative mod)
        float u = gx + 1.0f;
        u = u - 2.0f * floorf(u * 0.5f);
        gx = u - 1.0f;

        float ix = ((gx + 1.0f) * 64.0f - 1.0f) * 0.5f;
        float iy = ((gy + 1.0f) * 64.0f - 1.0f) * 0.5f;
        float ix0f = floorf(ix), iy0f = floorf(iy);
        float wx1 = ix - ix0f, wy1 = iy - iy0f;
        float wx0 = 1.0f - wx1, wy0 = 1.0f - wy1;
        int ix0 = (int)ix0f, iy0 = (int)iy0f;
        int ix1 = ix0 + 1,   iy1 = iy0 + 1;
        float vx0 = (ix0 >= 0 && ix0 < 64) ? 1.0f : 0.0f;
        float vx1 = (ix1 >= 0 && ix1 < 64) ? 1.0f : 0.0f;
        float vy0 = (iy0 >= 0 && iy0 < 64) ? 1.0f : 0.0f;
        float vy1 = (iy1 >= 0 && iy1 < 64) ? 1.0f : 0.0f;
        int cx0 = min(max(ix0, 0), 63), cx1 = min(max(ix1, 0), 63);
        int cy0 = min(max(iy0, 0), 63), cy1 = min(max(iy1, 0), 63);
        float w00 = wt * wx0 * wy0 * vx0 * vy0;
        float w10 = wt * wx1 * wy0 * vx1 * vy0;
        float w01 = wt * wx0 * wy1 * vx0 * vy1;
        float w11 = wt * wx1 * wy1 * vx1 * vy1;
        int o00 = cy0 * 64 + cx0, o10 = cy0 * 64 + cx1;
        int o01 = cy1 * 64 + cx0, o11 = cy1 * 64 + cx1;
        const float* imk = images + k * Cc * HW;
        #pragma unroll
        for (int c = 0; c < Cc; ++c) {
            const float* p = imk + c * HW;
            acc[c] += w00 * p[o00] + w10 * p[o10] + w01 * p[o01] + w11 * p[o11];
        }
    }
    #pragma unroll
    for (int c = 0; c < Cc; ++c)
        out[(t * Cc + c) * HW + pix] = acc[c];
}

// ---------------------------------------------------------------------------
extern "C" void kernel_launch(void* const* d_in, const int* in_sizes, int n_in,
                              void* d_out, int out_size, void* d_ws, size_t ws_size,
                              hipStream_t stream) {
    (void)in_sizes; (void)n_in; (void)out_size; (void)ws_size;
    const float* flows     = (const float*)d_in[0];  // (1,16,2,64,64)
    const float* images    = (const float*)d_in[1];  // (1,16,32,64,64)
    const float* decay_log = (const float*)d_in[2];  // scalar
    const float* w1        = (const float*)d_in[3];  // (16,64)
    const float* b1        = (const float*)d_in[4];  // (16,)
    const float* w2        = (const float*)d_in[5];  // (2,16)
    const float* b2        = (const float*)d_in[6];  // (2,)
    float* out = (float*)d_out;                      // (1,16,32,64,64)

    float* ws  = (float*)d_ws;
    float* cum = ws;                                 // 16*2*4096   = 131072 f
    float* A   = cum + Lt * 2 * HW;                  // 16*16*4096  = 1M f
    float* B   = A + Lt * HID * HW;                  // 16*16*4096  = 1M f

    cumsum_kernel<<<HW / 256, 256, 0, stream>>>(flows, cum);
    // 8192 tiles (2 parts * 16 frames * 256 col-tiles), 4 waves/block
    proj_kernel<<<2048, 128, 0, stream>>>(images, w1, b1, A, B);
    fuse_kernel<<<dim3(HW / 256, Lt), 256, 0, stream>>>(
        images, cum, A, B, decay_log, w2, b2, out);
}